// BatchedRasterLoss3D_1924145348886
// MI455X (gfx1250) — compile-verified
//
#include <hip/hip_runtime.h>
#include <hip/hip_bf16.h>

// ---------------- problem constants ----------------
#define RES   32
#define KZN   17            // RES/2+1
#define NF    (RES*RES*KZN) // 17408 rfftn modes
#define NSMP  2
#define NPTS  512
#define NE    1024
#define NTILE (NE/4)        // 256 WMMA tiles (4 elements x 4 vertices = 16 rows)

typedef __attribute__((ext_vector_type(2))) float v2f;
typedef __attribute__((ext_vector_type(8))) float v8f;

#define TWOPI 6.2831853071795864769f

// ---------------- small helpers ----------------
__device__ __forceinline__ float u01hash(unsigned x) {
    x ^= x >> 16; x *= 0x7feb352dU;
    x ^= x >> 15; x *= 0x846ca68bU;
    x ^= x >> 16;
    return (float)(x >> 8) * (1.0f / 16777216.0f);
}

// fetch jittered vertex coordinate (mirrors V64 = V + 1e-4*(u-0.5), hash-based PRNG)
__device__ __forceinline__ float vcoord(const float* V, int n, int p, int d) {
    float v = V[((size_t)n * NPTS + p) * 3 + d];
    float u = u01hash((unsigned)((n * NPTS + p) * 3 + d) * 0x9E3779B9u + 0x85EBCA77u);
    return v + 1e-4f * (u - 0.5f);
}

// low 32 bits of a flat shared-memory address == LDS byte offset (aperture mapping)
__device__ __forceinline__ unsigned lds_off_u32(const void* p) {
    return (unsigned)(uintptr_t)p;
}

// ---------------- kernel 0: zero accumulators ----------------
__global__ void zero_accum(float* out, float* sumcd) {
    if (threadIdx.x == 0) { out[0] = 0.0f; sumcd[0] = 0.0f; sumcd[1] = 0.0f; }
}

// ---------------- kernel 1: gather elements, det, pack WMMA-A fragments ----------------
// Apack layout: [n][tile][lane] float2. Row M (0..15) of a tile = element (tile*4+M/4), vertex t=M%4.
// Lane l<16 holds (x,y) for row l (K=0,1); lane l+16 holds (z,0) for row l (K=2,3).
__global__ __launch_bounds__(256) void build_elems(const float* V, const long long* E,
                                                   const float* D, float2* Apack,
                                                   float* cdt, float* sumcd) {
    int id = blockIdx.x * blockDim.x + threadIdx.x;
    if (id >= NSMP * NE) return;
    int n = id / NE, e = id % NE;

    float X[4][3];
    for (int t = 0; t < 4; ++t) {
        int p = (int)E[(size_t)e * 4 + t];
        for (int d = 0; d < 3; ++d) X[t][d] = vcoord(V, n, p, d);
    }
    // det([x0-x3, x1-x3, x2-x3])
    float a0 = X[0][0]-X[3][0], a1 = X[0][1]-X[3][1], a2 = X[0][2]-X[3][2];
    float b0 = X[1][0]-X[3][0], b1 = X[1][1]-X[3][1], b2 = X[1][2]-X[3][2];
    float c0 = X[2][0]-X[3][0], c1 = X[2][1]-X[3][1], c2 = X[2][2]-X[3][2];
    float det = a0*(b1*c2 - b2*c1) - a1*(b0*c2 - b2*c0) + a2*(b0*c1 - b1*c0);
    float cd  = det * D[e];

    cdt[(size_t)n * NE + e] = cd;
    atomicAdd(&sumcd[n], cd);

    int tile = e >> 2;
    int rbase = (e & 3) * 4;
    float2* Ap = Apack + ((size_t)n * NTILE + tile) * 32;
    for (int t = 0; t < 4; ++t) {
        int r = rbase + t;
        Ap[r]      = make_float2(X[t][0], X[t][1]); // K=0,1
        Ap[r + 16] = make_float2(X[t][2], 0.0f);    // K=2,3
    }
}

// ---------------- per-element NUFFT term ----------------
__device__ __forceinline__ void accum_term(float s0, float s1, float s2, float s3,
                                           float cd, float& Fre, float& Fim) {
    float sn0, c0v, sn1, c1v, sn2, c2v, sn3, c3v;
    __sincosf(s0, &sn0, &c0v); __sincosf(s1, &sn1, &c1v);
    __sincosf(s2, &sn2, &c2v); __sincosf(s3, &sn3, &c3v);
    float d01 = s0 - s1, d02 = s0 - s2, d03 = s0 - s3;
    float d12 = s1 - s2, d13 = s1 - s3, d23 = s2 - s3;
    float i0 = __builtin_amdgcn_rcpf(-(d01 * d02 * d03));
    float i1 = __builtin_amdgcn_rcpf(  d01 * d12 * d13 );
    float i2 = __builtin_amdgcn_rcpf(-(d02 * d12 * d23));
    float i3 = __builtin_amdgcn_rcpf(  d03 * d13 * d23 );
    // F += i*cd*Sum( (cos - i sin)/den ) -> Re += cd*Sum(sin/den), Im += cd*Sum(cos/den)
    Fre += cd * (sn0 * i0 + sn1 * i1 + sn2 * i2 + sn3 * i3);
    Fim += cd * (c0v * i0 + c1v * i1 + c2v * i2 + c3v * i3);
}

// ---------------- kernel 2: frequency accumulation (WMMA phases, async-LDS stage) ----------------
// grid: NSMP * ceil(NF/128) blocks of 256 threads (8 waves); each wave owns 16 modes.
#define BPS ((NF + 127) / 128)  // 136 blocks per sample
__global__ __launch_bounds__(256) void freq_accum(const float2* __restrict__ Apack,
                                                  const float2* __restrict__ cdt2,
                                                  const float*  __restrict__ sumcd,
                                                  float* __restrict__ Fk) {
    __shared__ float2 sA[NTILE * 32];   // 64 KB: this sample's packed A fragments

    const int lane = threadIdx.x & 31;
    const int wave = threadIdx.x >> 5;
    const int n     = blockIdx.x / BPS;
    const int fbase = (blockIdx.x % BPS) * 128 + wave * 16;
    const int col   = lane & 15;
    const int q     = fbase + col;
    const int qc    = (q < NF) ? q : 0;

    // ---- async global -> LDS stage of the 64 KB A-fragment array (ASYNCcnt path) ----
    {
        const char* gA = (const char*)(Apack + (size_t)n * NTILE * 32);
        unsigned base = lds_off_u32(&sA[0]);
        for (int i = threadIdx.x; i < (NTILE * 32 * 8) / 16; i += 256) {   // 4096 x 16B
            unsigned la = base + i * 16;
            const char* ga = gA + (size_t)i * 16;
            asm volatile("global_load_async_to_lds_b128 %0, %1, off"
                         :: "v"(la), "v"(ga) : "memory");
        }
        asm volatile("s_wait_asynccnt 0" ::: "memory");
        __syncthreads();
    }

    int iz = qc % KZN; int r = qc / KZN; int iy = r & 31; int ix = r >> 5;
    float kx = (ix < 16) ? (float)ix : (float)(ix - 32);
    float ky = (iy < 16) ? (float)iy : (float)(iy - 32);
    float kz = (float)iz;

    // B fragment: K rows = 2*pi*(kx,ky,kz,0); lanes 0-15: K=0,1; lanes 16-31: K=2,3
    v2f b;
    if (lane < 16) { b.x = TWOPI * kx; b.y = TWOPI * ky; }
    else           { b.x = TWOPI * kz; b.y = 0.0f;       }

    float Fre = 0.0f, Fim = 0.0f;
    const float2* cdp = cdt2 + (size_t)n * (NE / 2) + ((lane >> 4) & 1);

    #pragma unroll 4
    for (int tile = 0; tile < NTILE; ++tile) {
        __builtin_prefetch(cdp + (tile + 16) * 2, 0, 1);   // global_prefetch_b8
        float2 a2 = sA[tile * 32 + lane];                  // ds_load_b64, conflict-free
        v2f a; a.x = a2.x; a.y = a2.y;
        v8f c = {};
        // phases D[M,N] = 2*pi * k_N . x_M   (v_wmma_f32_16x16x4_f32)
        v8f d = __builtin_amdgcn_wmma_f32_16x16x4_f32(
            false, a, false, b, (short)0, c, false, false);
        float2 cd = cdp[(size_t)tile * 2];
        accum_term(d[0], d[1], d[2], d[3], cd.x, Fre, Fim); // element 0 (lanes<16) / 2
        accum_term(d[4], d[5], d[6], d[7], cd.y, Fre, Fim); // element 1 (lanes<16) / 3
    }

    // fold lane l and l+16 (same frequency column)
    Fre += __shfl_xor(Fre, 16, 32);
    Fim += __shfl_xor(Fim, 16, 32);

    if (lane < 16 && q < NF) {
        float dis2 = kx * kx + ky * ky + kz * kz;
        float g = __expf(-0.5f * (3.0f / 32.0f) * (3.0f / 32.0f) * dis2);
        float re = Fre * g, im = Fim * g;
        if (q == 0) { re = sumcd[n] * (1.0f / 6.0f); im = 0.0f; } // DC = total mass
        Fk[((size_t)n * NF + q) * 2 + 0] = re;
        Fk[((size_t)n * NF + q) * 2 + 1] = im;
    }
}

// ---------------- kernels 3-5: separable inverse rFFT (x RES^3 folded in) ----------------
__global__ __launch_bounds__(256) void ifft_x(const float* __restrict__ Fk, float* __restrict__ T1) {
    int id = blockIdx.x * blockDim.x + threadIdx.x;
    if (id >= NSMP * RES * RES * KZN) return;
    int kz = id % KZN; int r = id / KZN; int ky = r & 31; r >>= 5; int x = r & 31; int n = r >> 5;
    float re = 0.0f, im = 0.0f;
    for (int kx = 0; kx < RES; ++kx) {
        float ang = TWOPI * (float)((kx * x) & 31) * (1.0f / 32.0f);
        float sn, cs; __sincosf(ang, &sn, &cs);
        const float* p = Fk + ((((size_t)n * RES + kx) * RES + ky) * KZN + kz) * 2;
        re += p[0] * cs - p[1] * sn;
        im += p[0] * sn + p[1] * cs;
    }
    float* o = T1 + ((((size_t)n * RES + x) * RES + ky) * KZN + kz) * 2;
    o[0] = re; o[1] = im;
}

__global__ __launch_bounds__(256) void ifft_y(const float* __restrict__ T1, float* __restrict__ T2) {
    int id = blockIdx.x * blockDim.x + threadIdx.x;
    if (id >= NSMP * RES * RES * KZN) return;
    int kz = id % KZN; int r = id / KZN; int y = r & 31; r >>= 5; int x = r & 31; int n = r >> 5;
    float re = 0.0f, im = 0.0f;
    for (int ky = 0; ky < RES; ++ky) {
        float ang = TWOPI * (float)((ky * y) & 31) * (1.0f / 32.0f);
        float sn, cs; __sincosf(ang, &sn, &cs);
        const float* p = T1 + ((((size_t)n * RES + x) * RES + ky) * KZN + kz) * 2;
        re += p[0] * cs - p[1] * sn;
        im += p[0] * sn + p[1] * cs;
    }
    float* o = T2 + ((((size_t)n * RES + x) * RES + y) * KZN + kz) * 2;
    o[0] = re; o[1] = im;
}

__global__ __launch_bounds__(256) void ifft_z(const float* __restrict__ T2, float* __restrict__ f) {
    int id = blockIdx.x * blockDim.x + threadIdx.x;
    if (id >= NSMP * RES * RES * RES) return;
    int z = id & 31; int r = id >> 5; int y = r & 31; r >>= 5; int x = r & 31; int n = r >> 5;
    const float* p = T2 + (((size_t)n * RES + x) * RES + y) * KZN * 2;
    float acc = p[0];                                   // kz = 0 (real part)
    acc += ((z & 1) ? -1.0f : 1.0f) * p[16 * 2];        // kz = 16 Nyquist
    for (int kz = 1; kz < 16; ++kz) {
        float ang = TWOPI * (float)((kz * z) & 31) * (1.0f / 32.0f);
        float sn, cs; __sincosf(ang, &sn, &cs);
        acc += 2.0f * (p[kz * 2] * cs - p[kz * 2 + 1] * sn);
    }
    f[id] = acc;
}

// ---------------- kernel 6: L1 loss reduction ----------------
__global__ __launch_bounds__(256) void l1_loss(const float* __restrict__ f,
                                               const float* __restrict__ tgt,
                                               float* __restrict__ out) {
    __shared__ float sm[256];
    const int total = NSMP * RES * RES * RES;
    float s = 0.0f;
    for (int i = blockIdx.x * blockDim.x + threadIdx.x; i < total;
         i += gridDim.x * blockDim.x)
        s += fabsf(f[i] - tgt[i]);
    sm[threadIdx.x] = s;
    __syncthreads();
    for (int off = 128; off > 0; off >>= 1) {
        if (threadIdx.x < off) sm[threadIdx.x] += sm[threadIdx.x + off];
        __syncthreads();
    }
    if (threadIdx.x == 0)
        atomicAdd(out, sm[0] * (1.0f / (float)(NSMP * RES * RES * RES)));
}

// ---------------- launcher ----------------
extern "C" void kernel_launch(void* const* d_in, const int* in_sizes, int n_in,
                              void* d_out, int out_size, void* d_ws, size_t ws_size,
                              hipStream_t stream) {
    const float*     V   = (const float*)d_in[0];       // [2,512,3]
    const float*     tgt = (const float*)d_in[1];       // [2,32,32,32]
    const long long* E   = (const long long*)d_in[2];   // [1024,4] int64
    const float*     D   = (const float*)d_in[3];       // [1024,1]
    float* out = (float*)d_out;

    char* w = (char*)d_ws;
    float2* Apack = (float2*)(w);                      // 2*256*32 float2 = 131072 B
    float*  cdt   = (float*) (w + 131072);             // 2*1024 f32      =   8192 B
    float*  sumcd = (float*) (w + 139264);             // 2 f32 (pad 256)
    float*  Fk    = (float*) (w + 139520);             // 2*17408*2 f32   = 278528 B
    float*  T1    = (float*) (w + 418048);             // 278528 B
    float*  T2    = (float*) (w + 696576);             // 278528 B
    float*  f     = (float*) (w + 975104);             // 2*32768 f32     = 262144 B

    zero_accum<<<1, 32, 0, stream>>>(out, sumcd);

    build_elems<<<(NSMP * NE + 255) / 256, 256, 0, stream>>>(V, E, D, Apack, cdt, sumcd);

    freq_accum<<<NSMP * BPS, 256, 0, stream>>>(Apack, (const float2*)cdt, sumcd, Fk);

    int nlin = NSMP * RES * RES * KZN;                 // 34816
    ifft_x<<<(nlin + 255) / 256, 256, 0, stream>>>(Fk, T1);
    ifft_y<<<(nlin + 255) / 256, 256, 0, stream>>>(T1, T2);
    ifft_z<<<(NSMP * RES * RES * RES + 255) / 256, 256, 0, stream>>>(T2, f);

    l1_loss<<<64, 256, 0, stream>>>(f, tgt, out);
}